// GNNStack_1185410974147
// MI455X (gfx1250) — compile-verified
//
#include <hip/hip_runtime.h>
#include <hip/hip_bf16.h>
#include <math.h>

typedef float v2f __attribute__((ext_vector_type(2)));
typedef float v8f __attribute__((ext_vector_type(8)));

#define N_NODES 50000
#define N_EDGES 600000

// ---------------- utility kernels ----------------

__global__ void zero_kernel(float* __restrict__ p, int n) {
    int i = blockIdx.x * blockDim.x + threadIdx.x;
    if (i < n) p[i] = 0.0f;
}

__global__ void count_kernel(const long long* __restrict__ dst,
                             float* __restrict__ cnt, int E) {
    int e = blockIdx.x * blockDim.x + threadIdx.x;
    if (e < E) atomicAdd(&cnt[(int)dst[e]], 1.0f);
}

// scatter-add x[src] into agg[dst]; grid.y indexes 4-float chunks of the feature dim
__global__ void scatter_kernel(const float* __restrict__ X,
                               const long long* __restrict__ src,
                               const long long* __restrict__ dst,
                               float* __restrict__ agg, int E, int K) {
    int e = blockIdx.x * blockDim.x + threadIdx.x;
    if (e >= E) return;
    int c = blockIdx.y << 2;
    int s = (int)src[e];
    int d = (int)dst[e];
    float4 v = *(const float4*)(X + (long)s * K + c);
    float* p = agg + (long)d * K + c;
    atomicAdd(p + 0, v.x);
    atomicAdd(p + 1, v.y);
    atomicAdd(p + 2, v.z);
    atomicAdd(p + 3, v.w);
}

__global__ void mean_kernel(float* __restrict__ agg, const float* __restrict__ cnt,
                            int total, int K) {
    int i = blockIdx.x * blockDim.x + threadIdx.x;
    if (i >= total) return;
    float c = cnt[i / K];
    agg[i] = agg[i] / fmaxf(c, 1.0f);
}

// zero-pad layer-3 weights 40x256 -> 64x256 and bias 40 -> 64
__global__ void pad_w3_kernel(const float* __restrict__ Wl3,
                              const float* __restrict__ Wr3,
                              const float* __restrict__ bl3,
                              float* __restrict__ Wlp,
                              float* __restrict__ Wrp,
                              float* __restrict__ blp) {
    int i = blockIdx.x * blockDim.x + threadIdx.x;   // over 64*256
    if (i < 64 * 256) {
        int col = i >> 8;                            // row-major [col][k], same layout
        float vl = (col < 40) ? Wl3[i] : 0.0f;
        float vr = (col < 40) ? Wr3[i] : 0.0f;
        Wlp[i] = vl;
        Wrp[i] = vr;
    }
    if (i < 64) blp[i] = (i < 40) ? bl3[i] : 0.0f;
}

__global__ void log_softmax_kernel(float* __restrict__ Y, int M, int N) {
    int r = blockIdx.x * blockDim.x + threadIdx.x;
    if (r >= M) return;
    float* row = Y + (long)r * N;
    float mx = -INFINITY;
    for (int j = 0; j < N; ++j) mx = fmaxf(mx, row[j]);
    float s = 0.0f;
    for (int j = 0; j < N; ++j) s += expf(row[j] - mx);
    float lse = mx + logf(s);
    for (int j = 0; j < N; ++j) row[j] = row[j] - lse;
}

// ---------------- fused SAGE GEMM via fp32 WMMA ----------------
// Y[m][n] = sum_k Xmean[m][k]*Wl[n][k] + bl[n] + sum_k Xself[m][k]*Wr[n][k]
// Wave computes 16(M) x 64(N). Weights are padded so the K-loop has NO guards.
// A (16x4 f32) frag: {X[m0+(lane&15)][k+koff], [k+koff+1]}, koff = 2*(lane>=16)
// B (4x16 f32) frag: {W[n0+(lane&15)][k+koff], [k+koff+1]}
// D: VGPR g -> row m0 + g + 8*(lane>=16), col n0 + (lane&15)
template <int K, bool RELU>
__global__ void __launch_bounds__(256)
sage_gemm(const float* __restrict__ Xself, const float* __restrict__ Xmean,
          const float* __restrict__ Wl, const float* __restrict__ bl,
          const float* __restrict__ Wr, float* __restrict__ Y,
          int M, int Nout) {
    const int lane  = threadIdx.x & 31;
    const int wave  = threadIdx.x >> 5;
    const int mtile = blockIdx.x * 8 + wave;
    if (mtile * 16 >= M) return;                 // wave-uniform: EXEC stays all-ones

    const int m0   = mtile * 16;
    const int n0   = blockIdx.y * 64;
    const int lrow = lane & 15;
    const int koff = (lane >> 4) << 1;           // 0 or 2

    const float* pm = Xmean + (long)(m0 + lrow) * K + koff;
    const float* ps = Xself + (long)(m0 + lrow) * K + koff;
    const float* pl = Wl + (long)(n0 + lrow) * K + koff;
    const float* pr = Wr + (long)(n0 + lrow) * K + koff;

    v8f a0 = v8f{0.f,0.f,0.f,0.f,0.f,0.f,0.f,0.f};
    v8f a1 = a0, a2 = a0, a3 = a0;

    for (int k = 0; k < K; k += 4) {
        v2f am = *(const v2f*)pm;
        v2f as = *(const v2f*)ps;
        v2f l0 = *(const v2f*)(pl);
        v2f l1 = *(const v2f*)(pl + 16 * K);     // constant byte offsets -> IOFFSET
        v2f l2 = *(const v2f*)(pl + 32 * K);
        v2f l3 = *(const v2f*)(pl + 48 * K);
        v2f r0 = *(const v2f*)(pr);
        v2f r1 = *(const v2f*)(pr + 16 * K);
        v2f r2 = *(const v2f*)(pr + 32 * K);
        v2f r3 = *(const v2f*)(pr + 48 * K);

        a0 = __builtin_amdgcn_wmma_f32_16x16x4_f32(false, am, false, l0, (short)0, a0, false, false);
        a0 = __builtin_amdgcn_wmma_f32_16x16x4_f32(false, as, false, r0, (short)0, a0, false, false);
        a1 = __builtin_amdgcn_wmma_f32_16x16x4_f32(false, am, false, l1, (short)0, a1, false, false);
        a1 = __builtin_amdgcn_wmma_f32_16x16x4_f32(false, as, false, r1, (short)0, a1, false, false);
        a2 = __builtin_amdgcn_wmma_f32_16x16x4_f32(false, am, false, l2, (short)0, a2, false, false);
        a2 = __builtin_amdgcn_wmma_f32_16x16x4_f32(false, as, false, r2, (short)0, a2, false, false);
        a3 = __builtin_amdgcn_wmma_f32_16x16x4_f32(false, am, false, l3, (short)0, a3, false, false);
        a3 = __builtin_amdgcn_wmma_f32_16x16x4_f32(false, as, false, r3, (short)0, a3, false, false);

        pm += 4; ps += 4; pl += 4; pr += 4;
    }

    const int rbase = m0 + ((lane >> 4) << 3);   // m0 or m0+8
    const int colb  = n0 + lrow;

#define STORE_TILE(ACC, T)                                         \
    do {                                                           \
        int c = colb + (T) * 16;                                   \
        if (c < Nout) {                                            \
            float bias = bl[c];                                    \
            _Pragma("unroll")                                      \
            for (int g = 0; g < 8; ++g) {                          \
                float v = ACC[g] + bias;                           \
                if (RELU) v = fmaxf(v, 0.0f);                      \
                Y[(long)(rbase + g) * Nout + c] = v;               \
            }                                                      \
        }                                                          \
    } while (0)

    STORE_TILE(a0, 0);
    STORE_TILE(a1, 1);
    STORE_TILE(a2, 2);
    STORE_TILE(a3, 3);
#undef STORE_TILE
}

// ---------------- host launch ----------------

extern "C" void kernel_launch(void* const* d_in, const int* in_sizes, int n_in,
                              void* d_out, int out_size, void* d_ws, size_t ws_size,
                              hipStream_t stream) {
    const float*     x   = (const float*)d_in[0];
    const long long* ei  = (const long long*)d_in[1];
    const long long* src = ei;
    const long long* dst = ei + N_EDGES;
    const float* Wl1 = (const float*)d_in[2];
    const float* bl1 = (const float*)d_in[3];
    const float* Wr1 = (const float*)d_in[4];
    const float* Wl2 = (const float*)d_in[5];
    const float* bl2 = (const float*)d_in[6];
    const float* Wr2 = (const float*)d_in[7];
    const float* Wl3 = (const float*)d_in[8];
    const float* bl3 = (const float*)d_in[9];
    const float* Wr3 = (const float*)d_in[10];
    float* out = (float*)d_out;

    const int M = N_NODES, E = N_EDGES;
    float* ws    = (float*)d_ws;
    float* cnt   = ws;                            // M floats (200000 B, 16B aligned)
    float* meanb = ws + M;                        // M*256
    float* h1    = meanb + (long)M * 256;         // M*256
    float* h2    = h1 + (long)M * 256;            // M*256
    float* Wl3p  = h2 + (long)M * 256;            // 64*256
    float* Wr3p  = Wl3p + 64 * 256;               // 64*256
    float* bl3p  = Wr3p + 64 * 256;               // 64

    const int TPB = 256;
    dim3 eb((E + TPB - 1) / TPB);
    dim3 gemm_grid((M / 16 + 7) / 8, 4);          // 391 x 4
    dim3 gemm_grid3((M / 16 + 7) / 8, 1);         // 391 x 1

    // degree counts (shared by all layers) + padded layer-3 weights
    zero_kernel<<<(M + TPB - 1) / TPB, TPB, 0, stream>>>(cnt, M);
    count_kernel<<<eb, TPB, 0, stream>>>(dst, cnt, E);
    pad_w3_kernel<<<(64 * 256 + TPB - 1) / TPB, TPB, 0, stream>>>(Wl3, Wr3, bl3,
                                                                  Wl3p, Wr3p, bl3p);

    // ---- layer 1: 128 -> 256, relu ----
    {
        const int K = 128;
        int tot = M * K;
        zero_kernel<<<(tot + TPB - 1) / TPB, TPB, 0, stream>>>(meanb, tot);
        scatter_kernel<<<dim3(eb.x, K / 4), TPB, 0, stream>>>(x, src, dst, meanb, E, K);
        mean_kernel<<<(tot + TPB - 1) / TPB, TPB, 0, stream>>>(meanb, cnt, tot, K);
        sage_gemm<128, true><<<gemm_grid, TPB, 0, stream>>>(x, meanb, Wl1, bl1, Wr1,
                                                            h1, M, 256);
    }
    // ---- layer 2: 256 -> 256, relu ----
    {
        const int K = 256;
        int tot = M * K;
        zero_kernel<<<(tot + TPB - 1) / TPB, TPB, 0, stream>>>(meanb, tot);
        scatter_kernel<<<dim3(eb.x, K / 4), TPB, 0, stream>>>(h1, src, dst, meanb, E, K);
        mean_kernel<<<(tot + TPB - 1) / TPB, TPB, 0, stream>>>(meanb, cnt, tot, K);
        sage_gemm<256, true><<<gemm_grid, TPB, 0, stream>>>(h1, meanb, Wl2, bl2, Wr2,
                                                            h2, M, 256);
    }
    // ---- layer 3: 256 -> 40 (padded weights), then log_softmax ----
    {
        const int K = 256;
        int tot = M * K;
        zero_kernel<<<(tot + TPB - 1) / TPB, TPB, 0, stream>>>(meanb, tot);
        scatter_kernel<<<dim3(eb.x, K / 4), TPB, 0, stream>>>(h2, src, dst, meanb, E, K);
        mean_kernel<<<(tot + TPB - 1) / TPB, TPB, 0, stream>>>(meanb, cnt, tot, K);
        sage_gemm<256, false><<<gemm_grid3, TPB, 0, stream>>>(h2, meanb, Wl3p, bl3p,
                                                              Wr3p, out, M, 40);
        log_softmax_kernel<<<(M + TPB - 1) / TPB, TPB, 0, stream>>>(out, M, 40);
    }
}